// rnn_mp_hetero_44495861187264
// MI455X (gfx1250) — compile-verified
//
#include <hip/hip_runtime.h>

typedef __attribute__((ext_vector_type(2))) float v2f;
typedef __attribute__((ext_vector_type(8))) float v8f;

#define LDS_FENCE() asm volatile("" ::: "memory")

__device__ __forceinline__ float sigm(float x)  { return 1.0f / (1.0f + expf(-x)); }
__device__ __forceinline__ float lrelu(float x) { return x > 0.0f ? x : 0.01f * x; }

// ---------------- Kernel A: edge LSTM + segment_sum(he, row) ----------------
__global__ __launch_bounds__(256)
void k_edge_lstm(const float* __restrict__ eattr, const int* __restrict__ eidx,
                 const float* __restrict__ heh, const float* __restrict__ hec,
                 const float* __restrict__ Wih, const float* __restrict__ Whh,
                 const float* __restrict__ bih, const float* __restrict__ bhh,
                 float* __restrict__ out_he, float* __restrict__ out_ce,
                 float* __restrict__ edge_enc, int E)
{
  __shared__ float sWih[64];
  __shared__ float sWhh[256];
  __shared__ float sB[32];
  for (int i = threadIdx.x; i < 64;  i += blockDim.x) sWih[i] = Wih[i];
  for (int i = threadIdx.x; i < 256; i += blockDim.x) sWhh[i] = Whh[i];
  for (int i = threadIdx.x; i < 32;  i += blockDim.x) sB[i] = bih[i] + bhh[i];
  __syncthreads();
  int e = blockIdx.x * blockDim.x + threadIdx.x;
  if (e >= E) return;
  float x0 = eattr[2*(size_t)e], x1 = eattr[2*(size_t)e + 1];
  float h[8], c[8];
#pragma unroll
  for (int u = 0; u < 8; ++u) { h[u] = heh[8*(size_t)e + u]; c[u] = hec[8*(size_t)e + u]; }
  float g[32];
#pragma unroll
  for (int gi = 0; gi < 32; ++gi) {
    float s = sB[gi] + sWih[2*gi] * x0 + sWih[2*gi+1] * x1;
#pragma unroll
    for (int k = 0; k < 8; ++k) s += sWhh[8*gi + k] * h[k];
    g[gi] = s;
  }
  int r = eidx[e];
#pragma unroll
  for (int u = 0; u < 8; ++u) {
    float c2 = sigm(g[8+u]) * c[u] + sigm(g[u]) * tanhf(g[16+u]);
    float h2 = sigm(g[24+u]) * tanhf(c2);
    out_ce[8*(size_t)e + u] = c2;
    out_he[8*(size_t)e + u] = h2;
    atomicAdd(&edge_enc[8*(size_t)r + u], h2);
  }
}

// ---------------- Kernel B: node LSTM + build x_concat ----------------
__global__ __launch_bounds__(256)
void k_node_lstm(const float* __restrict__ x,
                 const float* __restrict__ hnh, const float* __restrict__ hnc,
                 const float* __restrict__ Wih, const float* __restrict__ Whh,
                 const float* __restrict__ bih, const float* __restrict__ bhh,
                 const float* __restrict__ edge_enc,
                 float* __restrict__ out_hn, float* __restrict__ out_cn,
                 float* __restrict__ xcat, int N)
{
  __shared__ float sWih[400];
  __shared__ float sWhh[1600];
  __shared__ float sB[80];
  for (int i = threadIdx.x; i < 400;  i += blockDim.x) sWih[i] = Wih[i];
  for (int i = threadIdx.x; i < 1600; i += blockDim.x) sWhh[i] = Whh[i];
  for (int i = threadIdx.x; i < 80;   i += blockDim.x) sB[i] = bih[i] + bhh[i];
  __syncthreads();
  int n = blockIdx.x * blockDim.x + threadIdx.x;
  if (n >= N) return;
  float xf[5];
#pragma unroll
  for (int j = 0; j < 5; ++j) xf[j] = x[10*(size_t)n + j];
  float h[20], c[20];
#pragma unroll
  for (int u = 0; u < 20; ++u) { h[u] = hnh[20*(size_t)n + u]; c[u] = hnc[20*(size_t)n + u]; }
  float g[80];
  for (int gi = 0; gi < 80; ++gi) {
    float s = sB[gi];
#pragma unroll
    for (int j = 0; j < 5; ++j) s += sWih[5*gi + j] * xf[j];
#pragma unroll
    for (int k = 0; k < 20; ++k) s += sWhh[20*gi + k] * h[k];
    g[gi] = s;
  }
#pragma unroll
  for (int u = 0; u < 20; ++u) {
    float c2 = sigm(g[20+u]) * c[u] + sigm(g[u]) * tanhf(g[40+u]);
    float h2 = sigm(g[60+u]) * tanhf(c2);
    out_cn[20*(size_t)n + u] = c2;
    out_hn[20*(size_t)n + u] = h2;
    xcat[28*(size_t)n + u] = h2;
  }
#pragma unroll
  for (int k = 0; k < 8; ++k) xcat[28*(size_t)n + 20 + k] = edge_enc[8*(size_t)n + k];
}

// ---------------- Kernel C: edge MLP (WMMA f32) + segment_sum(e_lat, row) ----------------
#define KC_WAVES 4
__global__ __launch_bounds__(32 * KC_WAVES)
void k_edge_mlp(const float* __restrict__ xcat, const int* __restrict__ eidx,
                const float* __restrict__ eattr,
                const float* __restrict__ W1, const float* __restrict__ b1,
                const float* __restrict__ W2, const float* __restrict__ b2,
                float* __restrict__ agg, int E)
{
  __shared__ float sW1[64 * 60];               // W1 zero-padded 58 -> 60 in K
  __shared__ float sW2[32 * 64];
  __shared__ float sT[KC_WAVES][16*60 + 16*64]; // per-wave: e_in tile + hidden tile
  for (int i = threadIdx.x; i < 64*60; i += blockDim.x) {
    int r = i / 60, cc = i % 60;
    sW1[i] = (cc < 58) ? W1[58*r + cc] : 0.0f;
  }
  for (int i = threadIdx.x; i < 32*64; i += blockDim.x) sW2[i] = W2[i];
  __syncthreads();

  const int wv   = threadIdx.x >> 5;
  const int lane = threadIdx.x & 31;
  const int tile = blockIdx.x * KC_WAVES + wv;
  const int e0   = tile * 16;
  float* ein = &sT[wv][0];
  float* hls = &sT[wv][16*60];

  // stage e_in = [xcat[row] | xcat[col] | edge_attr | 0pad] for 16 edges
  for (int i = lane; i < 16*60; i += 32) {
    int m = i / 60, j = i % 60;
    int e = e0 + m; if (e >= E) e = E - 1;
    float v;
    if (j < 28)      v = xcat[28*(size_t)eidx[e] + j];
    else if (j < 56) v = xcat[28*(size_t)eidx[(size_t)E + e] + (j - 28)];
    else if (j < 58) v = eattr[2*(size_t)e + (j - 56)];
    else             v = 0.0f;
    ein[60*m + j] = v;
  }
  LDS_FENCE();

  const int m      = lane & 15;
  const int off    = (lane < 16) ? 0 : 2;
  const int rowadd = (lane < 16) ? 0 : 8;

  // layer 1: (16x60) @ (60x64), bias carried in accumulator
  v8f acc[4];
#pragma unroll
  for (int nn = 0; nn < 4; ++nn) {
    float bv = b1[16*nn + m];
    v8f t = {bv,bv,bv,bv,bv,bv,bv,bv};
    acc[nn] = t;
  }
#pragma unroll
  for (int k = 0; k < 60; k += 4) {
    v2f a; a.x = ein[60*m + k + off]; a.y = ein[60*m + k + off + 1];
#pragma unroll
    for (int nn = 0; nn < 4; ++nn) {
      const int col = 16*nn + m;
      v2f b; b.x = sW1[60*col + k + off]; b.y = sW1[60*col + k + off + 1];
      acc[nn] = __builtin_amdgcn_wmma_f32_16x16x4_f32(false, a, false, b, (short)0, acc[nn], false, false);
    }
  }
  // leaky-relu + store hidden tile row-major to LDS
#pragma unroll
  for (int nn = 0; nn < 4; ++nn)
#pragma unroll
    for (int j = 0; j < 8; ++j)
      hls[64*(j + rowadd) + 16*nn + m] = lrelu(acc[nn][j]);
  LDS_FENCE();

  // layer 2: (16x64) @ (64x32)
  v8f a2[2];
#pragma unroll
  for (int nn = 0; nn < 2; ++nn) {
    float bv = b2[16*nn + m];
    v8f t = {bv,bv,bv,bv,bv,bv,bv,bv};
    a2[nn] = t;
  }
#pragma unroll
  for (int k = 0; k < 64; k += 4) {
    v2f a; a.x = hls[64*m + k + off]; a.y = hls[64*m + k + off + 1];
#pragma unroll
    for (int nn = 0; nn < 2; ++nn) {
      const int col = 16*nn + m;
      v2f b; b.x = sW2[64*col + k + off]; b.y = sW2[64*col + k + off + 1];
      a2[nn] = __builtin_amdgcn_wmma_f32_16x16x4_f32(false, a, false, b, (short)0, a2[nn], false, false);
    }
  }
  // scatter-add e_lat rows into agg[row]
#pragma unroll
  for (int j = 0; j < 8; ++j) {
    int e = e0 + j + rowadd;
    if (e < E) {
      int r = eidx[e];
#pragma unroll
      for (int nn = 0; nn < 2; ++nn)
        atomicAdd(&agg[32*(size_t)r + 16*nn + m], a2[nn][j]);
    }
  }
}

// ---------------- Kernel D: car MLP (WMMA f32) + type mask ----------------
#define KE_WAVES 2
__global__ __launch_bounds__(32 * KE_WAVES)
void k_car_mlp(const float* __restrict__ xcat, const float* __restrict__ agg,
               const float* __restrict__ x,
               const float* __restrict__ W1, const float* __restrict__ b1,
               const float* __restrict__ W2, const float* __restrict__ b2,
               const float* __restrict__ W3, const float* __restrict__ b3,
               float* __restrict__ out_y, int N)
{
  __shared__ float sW1[64 * 60];
  __shared__ float sW2[64 * 64];
  __shared__ float sW3[4 * 64];
  __shared__ float sT[KE_WAVES][16*60 + 16*64];
  for (int i = threadIdx.x; i < 64*60; i += blockDim.x) sW1[i] = W1[i];
  for (int i = threadIdx.x; i < 64*64; i += blockDim.x) sW2[i] = W2[i];
  for (int i = threadIdx.x; i < 4*64;  i += blockDim.x) sW3[i] = W3[i];
  __syncthreads();

  const int wv   = threadIdx.x >> 5;
  const int lane = threadIdx.x & 31;
  const int tile = blockIdx.x * KE_WAVES + wv;
  const int n0   = tile * 16;
  float* xin = &sT[wv][0];
  float* hls = &sT[wv][16*60];

  for (int i = lane; i < 16*60; i += 32) {
    int mm = i / 60, j = i % 60;
    int node = n0 + mm; if (node >= N) node = N - 1;
    float v = (j < 28) ? xcat[28*(size_t)node + j] : agg[32*(size_t)node + (j - 28)];
    xin[60*mm + j] = v;
  }
  LDS_FENCE();

  const int m      = lane & 15;
  const int off    = (lane < 16) ? 0 : 2;
  const int rowadd = (lane < 16) ? 0 : 8;

  // layer 1: (16x60) @ (60x64)
  v8f acc[4];
#pragma unroll
  for (int nn = 0; nn < 4; ++nn) {
    float bv = b1[16*nn + m];
    v8f t = {bv,bv,bv,bv,bv,bv,bv,bv};
    acc[nn] = t;
  }
#pragma unroll
  for (int k = 0; k < 60; k += 4) {
    v2f a; a.x = xin[60*m + k + off]; a.y = xin[60*m + k + off + 1];
#pragma unroll
    for (int nn = 0; nn < 4; ++nn) {
      const int col = 16*nn + m;
      v2f b; b.x = sW1[60*col + k + off]; b.y = sW1[60*col + k + off + 1];
      acc[nn] = __builtin_amdgcn_wmma_f32_16x16x4_f32(false, a, false, b, (short)0, acc[nn], false, false);
    }
  }
#pragma unroll
  for (int nn = 0; nn < 4; ++nn)
#pragma unroll
    for (int j = 0; j < 8; ++j)
      hls[64*(j + rowadd) + 16*nn + m] = lrelu(acc[nn][j]);
  LDS_FENCE();

  // layer 2: (16x64) @ (64x64)
  v8f acc2[4];
#pragma unroll
  for (int nn = 0; nn < 4; ++nn) {
    float bv = b2[16*nn + m];
    v8f t = {bv,bv,bv,bv,bv,bv,bv,bv};
    acc2[nn] = t;
  }
#pragma unroll
  for (int k = 0; k < 64; k += 4) {
    v2f a; a.x = hls[64*m + k + off]; a.y = hls[64*m + k + off + 1];
#pragma unroll
    for (int nn = 0; nn < 4; ++nn) {
      const int col = 16*nn + m;
      v2f b; b.x = sW2[64*col + k + off]; b.y = sW2[64*col + k + off + 1];
      acc2[nn] = __builtin_amdgcn_wmma_f32_16x16x4_f32(false, a, false, b, (short)0, acc2[nn], false, false);
    }
  }
  LDS_FENCE();   // all layer-2 reads of hls issued before overwrite (DS in-order per wave)
#pragma unroll
  for (int nn = 0; nn < 4; ++nn)
#pragma unroll
    for (int j = 0; j < 8; ++j)
      hls[64*(j + rowadd) + 16*nn + m] = lrelu(acc2[nn][j]);
  LDS_FENCE();

  // layer 3 (64 -> 4) via VALU + type mask; 64 outputs per tile, 2 per lane
#pragma unroll
  for (int t = 0; t < 2; ++t) {
    int oi = lane + 32*t;
    int mm = oi >> 2, cI = oi & 3;
    float s = b3[cI];
    for (int k = 0; k < 64; ++k) s += hls[64*mm + k] * sW3[64*cI + k];
    int node = n0 + mm;
    if (node < N) {
      bool any = false;
#pragma unroll
      for (int tt = 0; tt < 5; ++tt) any = any || (x[10*(size_t)node + 5 + tt] != 0.0f);
      out_y[4*(size_t)node + cI] = any ? s : 0.0f;
    }
  }
}

extern "C" void kernel_launch(void* const* d_in, const int* in_sizes, int n_in,
                              void* d_out, int out_size, void* d_ws, size_t ws_size,
                              hipStream_t stream)
{
  const float* x     = (const float*)d_in[0];
  const int*   eidx  = (const int*)  d_in[1];
  const float* eattr = (const float*)d_in[2];
  const float* hnh   = (const float*)d_in[3];
  const float* hnc   = (const float*)d_in[4];
  const float* heh   = (const float*)d_in[5];
  const float* hec   = (const float*)d_in[6];
  const float* nWih  = (const float*)d_in[7];
  const float* nWhh  = (const float*)d_in[8];
  const float* nbih  = (const float*)d_in[9];
  const float* nbhh  = (const float*)d_in[10];
  const float* eWih  = (const float*)d_in[11];
  const float* eWhh  = (const float*)d_in[12];
  const float* ebih  = (const float*)d_in[13];
  const float* ebhh  = (const float*)d_in[14];
  const float* eW1   = (const float*)d_in[15];
  const float* eb1   = (const float*)d_in[16];
  const float* eW2   = (const float*)d_in[17];
  const float* eb2   = (const float*)d_in[18];
  const float* cW1   = (const float*)d_in[19];
  const float* cb1   = (const float*)d_in[20];
  const float* cW2   = (const float*)d_in[21];
  const float* cb2   = (const float*)d_in[22];
  const float* cW3   = (const float*)d_in[23];
  const float* cb3   = (const float*)d_in[24];

  const int N = in_sizes[0] / 10;   // x is (N, 10)
  const int E = in_sizes[2] / 2;    // edge_attr is (E, 2)

  float* out_y  = (float*)d_out;              // N*4
  float* out_hn = out_y  + (size_t)N * 4;     // N*20
  float* out_cn = out_hn + (size_t)N * 20;    // N*20
  float* out_he = out_cn + (size_t)N * 20;    // E*8
  float* out_ce = out_he + (size_t)E * 8;     // E*8

  float* ws       = (float*)d_ws;
  float* edge_enc = ws;                       // N*8
  float* aggb     = ws + (size_t)N * 8;       // N*32
  float* xcat     = ws + (size_t)N * 40;      // N*28  (total ws: N*68 floats)

  // zero the accumulation buffers (edge_enc + agg are contiguous at ws start)
  hipMemsetAsync(d_ws, 0, (size_t)N * 40 * sizeof(float), stream);

  k_edge_lstm<<<(E + 255) / 256, 256, 0, stream>>>(eattr, eidx, heh, hec, eWih, eWhh,
                                                   ebih, ebhh, out_he, out_ce, edge_enc, E);
  k_node_lstm<<<(N + 255) / 256, 256, 0, stream>>>(x, hnh, hnc, nWih, nWhh, nbih, nbhh,
                                                   edge_enc, out_hn, out_cn, xcat, N);
  const int tilesE = (E + 15) / 16;
  k_edge_mlp<<<(tilesE + KC_WAVES - 1) / KC_WAVES, 32 * KC_WAVES, 0, stream>>>(
      xcat, eidx, eattr, eW1, eb1, eW2, eb2, aggb, E);
  const int tilesN = (N + 15) / 16;
  k_car_mlp<<<(tilesN + KE_WAVES - 1) / KE_WAVES, 32 * KE_WAVES, 0, stream>>>(
      xcat, aggb, x, cW1, cb1, cW2, cb2, cW3, cb3, out_y, N);
}